// RNN_19327352832373
// MI455X (gfx1250) — compile-verified
//
#include <hip/hip_runtime.h>
#include <math.h>

#define VOCAB 100000
#define EMB   128
#define HID   256
#define G3H   768   // 3*HID
#define SEQ   200

typedef float v2f __attribute__((ext_vector_type(2)));
typedef float v8f __attribute__((ext_vector_type(8)));

// ---------------------------------------------------------------------------
// Kernel A: xg[s][:] = embed[x[s]] @ w_ih^T + b_ih    (tiny: 200 x 768, K=128)
// ---------------------------------------------------------------------------
__global__ void k_embed_xg(const int* __restrict__ x,
                           const float* __restrict__ embed,
                           const float* __restrict__ w_ih,
                           const float* __restrict__ b_ih,
                           float* __restrict__ xg) {
  __shared__ float e[EMB];
  const int s   = blockIdx.x;
  const int tid = threadIdx.x;            // blockDim = 128
  e[tid] = embed[(size_t)x[s] * EMB + tid];
  __syncthreads();
  const float4* e4 = (const float4*)e;
  for (int j = tid; j < G3H; j += 128) {
    const float4* w4 = (const float4*)(w_ih + (size_t)j * EMB);
    float acc = b_ih[j];
#pragma unroll 8
    for (int k = 0; k < EMB / 4; ++k) {
      float4 wv = w4[k], ev = e4[k];
      acc += wv.x * ev.x + wv.y * ev.y + wv.z * ev.z + wv.w * ev.w;
    }
    xg[(size_t)s * G3H + j] = acc;
  }
}

// ---------------------------------------------------------------------------
// Kernel B: sequential GRU scan. One workgroup (768 threads = 24 wave32s on
// one WGP); h lives in LDS; 200 serial steps, two barriers/step.
// ---------------------------------------------------------------------------
__global__ void k_gru(const float* __restrict__ xg,
                      const float* __restrict__ w_hh,
                      const float* __restrict__ b_hh,
                      float* __restrict__ hs) {
  __shared__ float h[HID];
  __shared__ float gh[G3H];
  const int j = threadIdx.x;              // blockDim = 768
  if (j < HID) h[j] = 0.f;
  __syncthreads();
  const float4* w4 = (const float4*)(w_hh + (size_t)j * HID);
  const float bj = b_hh[j];
  for (int t = 0; t < SEQ; ++t) {
    const float4* h4 = (const float4*)h;
    float acc = bj;
#pragma unroll 8
    for (int k = 0; k < HID / 4; ++k) {
      float4 wv = w4[k], hv = h4[k];
      acc += wv.x * hv.x + wv.y * hv.y + wv.z * hv.z + wv.w * hv.w;
    }
    gh[j] = acc;
    __syncthreads();
    if (j < HID) {
      const float* xt = xg + (size_t)t * G3H;
      float r  = 1.f / (1.f + expf(-(xt[j] + gh[j])));
      float z  = 1.f / (1.f + expf(-(xt[HID + j] + gh[HID + j])));
      float n  = tanhf(xt[2 * HID + j] + r * gh[2 * HID + j]);
      float hn = (1.f - z) * n + z * h[j];
      h[j] = hn;
      hs[(size_t)t * HID + j] = hn;
    }
    __syncthreads();
  }
}

// ---------------------------------------------------------------------------
// Kernel C: logits = hs @ w_out^T + b_out via V_WMMA_F32_16X16X4_F32.
// Block = 128 threads (4 waves). Each block: MT=2 M-tiles (32 rows) x 64 cols
// (16 per wave). B fragment (w_out stream) reused across both M tiles ->
// per k-step: 1 global_load_b64, 2 ds_load_b64, 2 wmma. Branchless B loads
// (clamped row), masking only at store, so EXEC stays all-1s in the loop.
// A tile in LDS, row stride 260 dwords (stride%64=4 -> conflict-free).
// ---------------------------------------------------------------------------
#define AS_STRIDE 260
#define MT 2

__global__ void k_logits_wmma(const float* __restrict__ hs,
                              const float* __restrict__ w_out,
                              const float* __restrict__ b_out,
                              float* __restrict__ out) {
  __shared__ float As[MT * 16 * AS_STRIDE];
  const int tid  = threadIdx.x;           // 128
  const int lane = tid & 31;
  const int wid  = tid >> 5;              // 0..3
  const int m0   = blockIdx.y * (MT * 16);
  const int nblk = blockIdx.x * 64;

  // Stage A tile (32 x 256 f32) with b128 loads/stores; zero-pad past SEQ.
  for (int idx = tid; idx < MT * 16 * (HID / 4); idx += 128) {
    int r  = idx >> 6;                    // / (HID/4)
    int k4 = idx & 63;
    int m  = m0 + r;
    float4 v = {0.f, 0.f, 0.f, 0.f};
    if (m < SEQ) v = *(const float4*)(hs + (size_t)m * HID + 4 * k4);
    *(float4*)(As + r * AS_STRIDE + 4 * k4) = v;
  }
  __syncthreads();

  const int  n      = nblk + wid * 16 + (lane & 15);
  const bool nvalid = (n < VOCAB);
  const int  nc     = nvalid ? n : (VOCAB - 1);      // clamped, always in-bounds
  const float* wrow = w_out + (size_t)nc * HID;
  const int  koff   = (lane >> 4) * 2;    // low half-wave: K+0/1, high: K+2/3
  const float* arow0 = As + (lane & 15) * AS_STRIDE;
  const float* arow1 = arow0 + 16 * AS_STRIDE;

  v8f c0 = {0.f, 0.f, 0.f, 0.f, 0.f, 0.f, 0.f, 0.f};
  v8f c1 = {0.f, 0.f, 0.f, 0.f, 0.f, 0.f, 0.f, 0.f};

  for (int kb = 0; kb < HID; kb += 64) {
    // speculative prefetch of the next 256B chunk of this w_out row
    __builtin_prefetch(wrow + kb + 64, 0, 0);
#pragma unroll
    for (int k = kb; k < kb + 64; k += 4) {
      v2f b  = *(const v2f*)(wrow + k + koff);   // global, 8B, branchless
      v2f a0 = *(const v2f*)(arow0 + k + koff);  // LDS, 8B
      v2f a1 = *(const v2f*)(arow1 + k + koff);  // LDS, 8B
      // 8 args: (neg_a, A, neg_b, B, c_mod, C, reuse_a, reuse_b)
      c0 = __builtin_amdgcn_wmma_f32_16x16x4_f32(false, a0, false, b,
                                                 (short)0, c0, false, false);
      c1 = __builtin_amdgcn_wmma_f32_16x16x4_f32(false, a1, false, b,
                                                 (short)0, c1, false, false);
    }
  }

  // D layout: VGPR r -> row m0 + r + 8*(lane>>4) (+16 for tile 1), col n.
  const float bias  = nvalid ? b_out[n] : 0.f;
  const int   mbase = m0 + (lane >> 4) * 8;
#pragma unroll
  for (int r = 0; r < 8; ++r) {
    int m = mbase + r;
    if (nvalid && m < SEQ)
      out[(size_t)m * VOCAB + n] = c0[r] + bias;
  }
#pragma unroll
  for (int r = 0; r < 8; ++r) {
    int m = mbase + 16 + r;
    if (nvalid && m < SEQ)
      out[(size_t)m * VOCAB + n] = c1[r] + bias;
  }
}

// ---------------------------------------------------------------------------
// Kernel D: in-place log_softmax over each row of [SEQ, VOCAB].
// ---------------------------------------------------------------------------
__global__ void k_logsoftmax(float* __restrict__ out) {
  __shared__ float red[512];
  const int row = blockIdx.x;
  const int tid = threadIdx.x;            // 512
  float* p = out + (size_t)row * VOCAB;

  float m = -INFINITY;
  for (int i = tid; i < VOCAB; i += 512) m = fmaxf(m, p[i]);
  red[tid] = m;
  __syncthreads();
  for (int s = 256; s > 0; s >>= 1) {
    if (tid < s) red[tid] = fmaxf(red[tid], red[tid + s]);
    __syncthreads();
  }
  const float rowmax = red[0];
  __syncthreads();

  float sum = 0.f;
  for (int i = tid; i < VOCAB; i += 512) sum += expf(p[i] - rowmax);
  red[tid] = sum;
  __syncthreads();
  for (int s = 256; s > 0; s >>= 1) {
    if (tid < s) red[tid] += red[tid + s];
    __syncthreads();
  }
  const float logZ = rowmax + logf(red[0]);
  __syncthreads();

  for (int i = tid; i < VOCAB; i += 512) p[i] -= logZ;
}

// ---------------------------------------------------------------------------
extern "C" void kernel_launch(void* const* d_in, const int* in_sizes, int n_in,
                              void* d_out, int out_size, void* d_ws, size_t ws_size,
                              hipStream_t stream) {
  const int*   x     = (const int*)d_in[0];
  const float* embed = (const float*)d_in[1];
  const float* w_ih  = (const float*)d_in[2];
  const float* w_hh  = (const float*)d_in[3];
  const float* b_ih  = (const float*)d_in[4];
  const float* b_hh  = (const float*)d_in[5];
  const float* w_out = (const float*)d_in[6];
  const float* b_out = (const float*)d_in[7];
  float* out = (float*)d_out;

  float* xg = (float*)d_ws;               // [SEQ, 768]
  float* hs = xg + (size_t)SEQ * G3H;     // [SEQ, 256]

  k_embed_xg<<<SEQ, 128, 0, stream>>>(x, embed, w_ih, b_ih, xg);
  k_gru<<<1, G3H, 0, stream>>>(xg, w_hh, b_hh, hs);

  dim3 grid((VOCAB + 63) / 64, (SEQ + MT * 16 - 1) / (MT * 16));  // 1563 x 7
  k_logits_wmma<<<grid, 128, 0, stream>>>(hs, w_out, b_out, out);

  k_logsoftmax<<<SEQ, 512, 0, stream>>>(out);
}